// GIN_78606491452618
// MI455X (gfx1250) — compile-verified
//
#include <hip/hip_runtime.h>
#include <math.h>

typedef float v2f __attribute__((ext_vector_type(2)));
typedef float v8f __attribute__((ext_vector_type(8)));

#define DHC 128
#define POOLW 1792

// ---------------------------------------------------------------- zero fill
__global__ void zero_kernel(float* __restrict__ p, int n) {
    int i = blockIdx.x * blockDim.x + threadIdx.x;
    if (i < n) p[i] = 0.0f;
}

// ---------------------------------------------------------- CSR construction
__global__ void count_deg(const int* __restrict__ dst, int* __restrict__ deg, int E) {
    int e = blockIdx.x * blockDim.x + threadIdx.x;
    if (e < E) atomicAdd(&deg[dst[e]], 1);
}

// single-block exclusive scan (1024 threads, tiled Hillis-Steele with carry)
__global__ void scan_kernel(const int* __restrict__ deg, int* __restrict__ rowptr,
                            int* __restrict__ cursor, int n) {
    __shared__ int tile[1024];
    __shared__ int carry_s;
    if (threadIdx.x == 0) carry_s = 0;
    __syncthreads();
    for (int base = 0; base < n; base += 1024) {
        int i = base + threadIdx.x;
        int v = (i < n) ? deg[i] : 0;
        tile[threadIdx.x] = v;
        __syncthreads();
        for (int off = 1; off < 1024; off <<= 1) {
            int t = (threadIdx.x >= off) ? tile[threadIdx.x - off] : 0;
            __syncthreads();
            tile[threadIdx.x] += t;
            __syncthreads();
        }
        int carry = carry_s;
        int excl = carry + tile[threadIdx.x] - v;
        if (i < n) { rowptr[i] = excl; cursor[i] = excl; }
        __syncthreads();
        if (threadIdx.x == 0) carry_s = carry + tile[1023];
        __syncthreads();
    }
    if (threadIdx.x == 0) rowptr[n] = carry_s;
}

__global__ void csr_fill(const int* __restrict__ src, const int* __restrict__ dst,
                         int* __restrict__ cursor, int* __restrict__ srclist, int E) {
    int e = blockIdx.x * blockDim.x + threadIdx.x;
    if (e < E) {
        int slot = atomicAdd(&cursor[dst[e]], 1);
        srclist[slot] = src[e];
    }
}

// ------------------------------------------------- GIN aggregation (gather)
// one block (128 threads) per node; src ids staged in LDS; coalesced row reads
__global__ void gather_agg(const float* __restrict__ h, const int* __restrict__ rowptr,
                           const int* __restrict__ srclist, float* __restrict__ agg) {
    __shared__ int sl[128];
    const int node = blockIdx.x;
    const int c = threadIdx.x;
    const int s0 = rowptr[node], s1 = rowptr[node + 1];
    float acc = 0.0f;
    for (int base = s0; base < s1; base += 128) {
        int j = base + c;
        sl[c] = (j < s1) ? srclist[j] : 0;
        __syncthreads();
        int cnt = min(128, s1 - base);
        for (int k = 0; k < cnt; ++k)
            acc += h[(size_t)sl[k] * DHC + c];
        __syncthreads();
    }
    agg[(size_t)node * DHC + c] = acc;
}

// ------------------------------------------------------------- BN stats -> affine
__global__ void bn_stats(const float* __restrict__ colsum,
                         const float* __restrict__ colsumsq,
                         const float* __restrict__ gamma,
                         const float* __restrict__ beta,
                         float* __restrict__ scale,
                         float* __restrict__ shift, float invM) {
    int c = threadIdx.x;  // 128 threads
    float mean = colsum[c] * invM;
    float var  = colsumsq[c] * invM - mean * mean;
    float rstd = rsqrtf(var + 1e-5f);
    float sc = gamma[c] * rstd;
    scale[c] = sc;
    shift[c] = beta[c] - mean * sc;
}

// ---------------------------------------------------------------- WMMA GEMM
// MODE 0: A := A + A2             ; out = A@W + bias  ; accumulate col sum/sumsq
// MODE 1: A := relu(scale*A+shift); out = relu(A@W+b) ; fused segment add/max pool
// MODE 2: A plain                 ; out = relu(A@W+b)
// block = 256 threads (8 waves); block tile 64(M) x 128(N); wave tile 64 x 16.
template <int MODE>
__global__ void wmma_gemm(const float* __restrict__ A, const float* __restrict__ A2,
                          const float* __restrict__ scale, const float* __restrict__ shift,
                          const float* __restrict__ W, const float* __restrict__ bias,
                          float* __restrict__ out, int M, int K, int N,
                          float* __restrict__ colsum, float* __restrict__ colsumsq,
                          const int* __restrict__ batch, float* __restrict__ pooled,
                          int addOff, int maxOff) {
    __shared__ float At[64 * 128];   // A tile (transformed), 32 KB
    __shared__ float Wt[32 * 128];   // W sub-tile, 16 KB

    const int tid  = threadIdx.x;
    const int lane = tid & 31;
    const int wv   = tid >> 5;
    const int l15  = lane & 15;
    const int half = lane >> 4;
    const int m0   = blockIdx.x * 64;
    const int nb   = blockIdx.y * 128;

    v8f acc[4];
#pragma unroll
    for (int s = 0; s < 4; ++s)
#pragma unroll
        for (int e = 0; e < 8; ++e) acc[s][e] = 0.0f;

    for (int kc = 0; kc < K; kc += 128) {
        __syncthreads();
        // ---- stage A tile (64 rows x 128 K) with fused per-mode transform
        for (int i = tid; i < 64 * 32; i += 256) {
            int row = (i * 4) >> 7;
            int col = (i * 4) & 127;
            float4 v;
            int grow = m0 + row;
            if (grow < M) {
                v = *(const float4*)(A + (size_t)grow * K + kc + col);
                if (MODE == 0) {
                    float4 u = *(const float4*)(A2 + (size_t)grow * K + kc + col);
                    v.x += u.x; v.y += u.y; v.z += u.z; v.w += u.w;
                } else if (MODE == 1) {
                    int kg = kc + col;
                    v.x = fmaxf(fmaf(v.x, scale[kg + 0], shift[kg + 0]), 0.0f);
                    v.y = fmaxf(fmaf(v.y, scale[kg + 1], shift[kg + 1]), 0.0f);
                    v.z = fmaxf(fmaf(v.z, scale[kg + 2], shift[kg + 2]), 0.0f);
                    v.w = fmaxf(fmaf(v.w, scale[kg + 3], shift[kg + 3]), 0.0f);
                }
            } else {
                v = make_float4(0.0f, 0.0f, 0.0f, 0.0f);
            }
            *(float4*)&At[row * 128 + col] = v;
        }
        // ---- K sub-chunks of 32
        for (int ks = 0; ks < 128; ks += 32) {
            __syncthreads();
            for (int i = tid; i < 32 * 32; i += 256) {
                int row = (i * 4) >> 7;
                int col = (i * 4) & 127;
                *(float4*)&Wt[row * 128 + col] =
                    *(const float4*)(W + (size_t)(kc + ks + row) * N + nb + col);
            }
            __syncthreads();
#pragma unroll
            for (int kk = 0; kk < 32; kk += 4) {
                const int krow = kk + half * 2;  // A/B K placement per ISA layout
                v2f bfrag;
                bfrag.x = Wt[krow * 128 + (wv << 4) + l15];
                bfrag.y = Wt[(krow + 1) * 128 + (wv << 4) + l15];
#pragma unroll
                for (int s = 0; s < 4; ++s) {
                    v2f afrag = *(const v2f*)&At[(s * 16 + l15) * 128 + krow];
                    acc[s] = __builtin_amdgcn_wmma_f32_16x16x4_f32(
                        false, afrag, false, bfrag, (short)0, acc[s], false, false);
                }
            }
        }
    }

    // ---------------------------------------------------------- epilogue
    const int ncol = nb + (wv << 4) + l15;
    const float bv = bias[ncol];
    float s1 = 0.0f, s2 = 0.0f;
    // MODE 1: run-length pooling state (batch is sorted; rows visited in
    // increasing order per lane, so equal graph ids are contiguous)
    int curG = -1;
    float aadd = 0.0f, amax = 0.0f;
#pragma unroll
    for (int s = 0; s < 4; ++s) {
#pragma unroll
        for (int r = 0; r < 8; ++r) {
            int grow = m0 + s * 16 + ((lane < 16) ? r : (r + 8));
            if (grow < M) {
                float val = acc[s][r] + bv;
                if (MODE != 0) val = fmaxf(val, 0.0f);
                out[(size_t)grow * N + ncol] = val;
                if (MODE == 0) { s1 += val; s2 += val * val; }
                if (MODE == 1) {
                    int g = batch[grow];
                    if (g != curG) {
                        if (curG >= 0) {
                            float* pb = pooled + (size_t)curG * POOLW;
                            atomicAdd(pb + addOff + ncol, aadd);
                            atomicMax((unsigned int*)(pb + maxOff + ncol),
                                      __float_as_uint(amax));
                        }
                        curG = g; aadd = 0.0f; amax = 0.0f;
                    }
                    aadd += val;
                    amax = fmaxf(amax, val);
                }
            }
        }
    }
    if (MODE == 0) {
        // lanes l and l+16 hold the same output column: fold, then 16 atomics/wave
        s1 += __shfl_xor(s1, 16, 32);
        s2 += __shfl_xor(s2, 16, 32);
        if (lane < 16) {
            atomicAdd(&colsum[ncol], s1);
            atomicAdd(&colsumsq[ncol], s2);
        }
    }
    if (MODE == 1 && curG >= 0) {
        float* pb = pooled + (size_t)curG * POOLW;
        atomicAdd(pb + addOff + ncol, aadd);
        atomicMax((unsigned int*)(pb + maxOff + ncol), __float_as_uint(amax));
    }
}

// ---------------------------------------------- final dot + bias + sigmoid
__global__ void final_kernel(const float* __restrict__ t, const float* __restrict__ Wl2,
                             const float* __restrict__ bl2, float* __restrict__ outp,
                             int G, int K) {
    int g = blockIdx.x * blockDim.x + threadIdx.x;
    if (g >= G) return;
    const float* row = t + (size_t)g * K;
    float acc = 0.0f;
    for (int k = 0; k < K; ++k) acc = fmaf(row[k], Wl2[k], acc);
    float logit = acc + bl2[0];
    outp[g] = 1.0f / (1.0f + expf(-logit));
    outp[G + g] = logit;
}

extern "C" void kernel_launch(void* const* d_in, const int* in_sizes, int n_in,
                              void* d_out, int out_size, void* d_ws, size_t ws_size,
                              hipStream_t stream) {
    const float* x     = (const float*)d_in[0];
    const int*   ei    = (const int*)d_in[1];
    const int*   batch = (const int*)d_in[2];
    const float* W1    = (const float*)d_in[3];
    const float* b1    = (const float*)d_in[4];
    const float* gamma = (const float*)d_in[5];
    const float* beta  = (const float*)d_in[6];
    const float* W2    = (const float*)d_in[7];
    const float* b2    = (const float*)d_in[8];
    const float* Wl1   = (const float*)d_in[9];
    const float* bl1   = (const float*)d_in[10];
    const float* Wl2   = (const float*)d_in[11];
    const float* bl2   = (const float*)d_in[12];

    const int Nn = in_sizes[0] / DHC;   // 50000
    const int E  = in_sizes[1] / 2;     // 800000
    const int G  = out_size / 2;        // 256

    float* ws      = (float*)d_ws;
    float* buffA   = ws;                               // h      [Nn*128]
    float* buffB   = buffA + (size_t)Nn * DHC;         // agg/z  [Nn*128]
    float* pooled  = buffB + (size_t)Nn * DHC;         // [G*1792]
    float* tbuf    = pooled + (size_t)G * POOLW;       // [G*1792]
    float* smallb  = tbuf + (size_t)G * POOLW;         // 512 floats
    float* colsum = smallb, *colsumsq = smallb + 128, *scale = smallb + 256,
          *shift = smallb + 384;
    int* ibase   = (int*)(smallb + 512);
    int* deg     = ibase;                // [Nn]   (also zero-init target)
    int* rowptr  = deg + Nn;             // [Nn+1]
    int* cursor  = rowptr + Nn + 1;      // [Nn]
    int* srclist = cursor + Nn;          // [E]

    const int NT = 256;
    const int* esrc = ei;
    const int* edst = ei + E;

    // ---- build CSR (dst -> list of src) once per launch
    zero_kernel<<<(Nn + NT - 1) / NT, NT, 0, stream>>>((float*)deg, Nn);
    count_deg<<<(E + NT - 1) / NT, NT, 0, stream>>>(edst, deg, E);
    scan_kernel<<<1, 1024, 0, stream>>>(deg, rowptr, cursor, Nn);
    csr_fill<<<(E + NT - 1) / NT, NT, 0, stream>>>(esrc, edst, cursor, srclist, E);

    zero_kernel<<<(G * POOLW + NT - 1) / NT, NT, 0, stream>>>(pooled, G * POOLW);

    const dim3 gNode((Nn + 63) / 64, 1);
    for (int i = 0; i < 7; ++i) {
        const float* h = (i == 0) ? x : buffA;
        zero_kernel<<<1, NT, 0, stream>>>(smallb, 256);
        // agg[v] = sum_{(u->v)} h[u]   (pure gather, no atomics)
        gather_agg<<<Nn, 128, 0, stream>>>(h, rowptr, srclist, buffB);
        // z = (agg + h) @ W1 + b1  (in-place over agg) + column stats
        wmma_gemm<0><<<gNode, NT, 0, stream>>>(
            buffB, h, nullptr, nullptr, W1 + (size_t)i * DHC * DHC, b1 + i * DHC,
            buffB, Nn, DHC, DHC, colsum, colsumsq, nullptr, nullptr, 0, 0);
        bn_stats<<<1, 128, 0, stream>>>(colsum, colsumsq, gamma + i * DHC,
                                        beta + i * DHC, scale, shift, 1.0f / Nn);
        // h' = relu( relu(BN(z)) @ W2 + b2 ) + fused add/max pooling
        wmma_gemm<1><<<gNode, NT, 0, stream>>>(
            buffB, nullptr, scale, shift, W2 + (size_t)i * DHC * DHC, b2 + i * DHC,
            buffA, Nn, DHC, DHC, nullptr, nullptr, batch, pooled,
            i * DHC, (7 + i) * DHC);
    }

    // t = relu(pooled @ Wl1 + bl1)   (256 x 1792 x 1792)
    const dim3 gMLP(G / 64, POOLW / 128);
    wmma_gemm<2><<<gMLP, NT, 0, stream>>>(
        pooled, nullptr, nullptr, nullptr, Wl1, bl1, tbuf, G, POOLW, POOLW,
        nullptr, nullptr, nullptr, nullptr, 0, 0);

    final_kernel<<<1, NT, 0, stream>>>(tbuf, Wl2, bl2, (float*)d_out, G, POOLW);
}